// SpikingVestibular_59777354826370
// MI455X (gfx1250) — compile-verified
//
#include <hip/hip_runtime.h>

// ---------------------------------------------------------------------------
// SpikingVestibular on MI455X (gfx1250): B=500k rows, N=6 neurons,
// 10 Izhikevich steps + closed 2-compartment relaxation.
//
// Roofline: ~176 MB total traffic (120 MB noise + 36 MB state + 12 MB out)
// -> ~7.6 us at 23.3 TB/s; ~0.4 GFLOP of VALU work is negligible. Pure
// HBM-streaming kernel. The CDNA5-specific win: stage the noise stream with
// GLOBAL_LOAD_ASYNC_TO_LDS (ASYNCcnt path). Each wave issues all 30 async
// b64 copies up front (in-order completion), then drains step s after
// s_wait_asynccnt (27 - 3s). 30 outstanding ops/wave at near-zero VGPR cost
// keeps HBM saturated without relying on occupancy alone.
// ---------------------------------------------------------------------------

typedef float f32x2 __attribute__((ext_vector_type(2)));
typedef int   v2i   __attribute__((vector_size(8)));   // matches builtin param

#define T_STEPS 10
#define T_N 6
#define THREADS 256

#define GLOBAL_AS __attribute__((address_space(1)))
#define LDS_AS    __attribute__((address_space(3)))

// Wave-scalar wait on the async counter; "memory" clobber stops the compiler
// from hoisting the LDS drain loads above the wait.
#define WAIT_ASYNCCNT(n) asm volatile("s_wait_asynccnt " #n ::: "memory")

// One per-lane 8-byte async copy global -> LDS (tracked by ASYNCcnt).
__device__ __forceinline__ void async_copy_b64(const float* gsrc, float* lds_dst) {
#if __has_builtin(__builtin_amdgcn_global_load_async_to_lds_b64)
    __builtin_amdgcn_global_load_async_to_lds_b64(
        (GLOBAL_AS v2i*)gsrc, (LDS_AS v2i*)lds_dst, /*offset=*/0, /*cpol=*/0);
#else
    unsigned lds_off = (unsigned)(size_t)(LDS_AS void*)lds_dst;
    asm volatile("global_load_async_to_lds_b64 %0, %1, off"
                 :: "v"(lds_off), "v"(gsrc) : "memory");
#endif
}

__global__ __launch_bounds__(THREADS) void spiking_vestibular_kernel(
    const float* __restrict__ speed_p,
    const float* __restrict__ turn_p,
    const float* __restrict__ pturn_p,
    const float* __restrict__ pspeed_p,
    const float* __restrict__ noise_p,
    const float* __restrict__ v0_p,
    const float* __restrict__ u0_p,
    const float* __restrict__ rate0_p,
    float* __restrict__ out_p,
    int B)
{
    // 10 steps x 256 threads x 6 floats = 60 KB -> 5 blocks / 320 KB WGP.
    __shared__ float nbuf[T_STEPS][THREADS][T_N];

    const int tid = threadIdx.x;
    int b = blockIdx.x * THREADS + tid;
    const bool active = (b < B);
    if (!active) b = B - 1;          // clamp: EXEC stays all-ones for async ops

    const size_t base        = (size_t)b * T_N;   // 24B row stride, 8B aligned
    const size_t step_stride = (size_t)B * T_N;

    // ---- Issue ALL noise copies for all 10 steps (30 async b64 per lane) ----
    #pragma unroll
    for (int s = 0; s < T_STEPS; ++s) {
        const float* np = noise_p + (size_t)s * step_stride + base;
        #pragma unroll
        for (int j = 0; j < 3; ++j)
            async_copy_b64(np + 2 * j, &nbuf[s][tid][2 * j]);
    }

    // ---- Overlap: scalar inputs + state while the DMA stream is in flight ---
    const float speed  = speed_p[b];
    const float turn   = turn_p[b];
    const float pturn  = pturn_p[b];
    const float pspeed = pspeed_p[b];

    const float tilt = fminf(1.0f, fabsf(turn) * speed * 0.5f);

    float I[T_N];
    I[0] = fmaxf(0.0f,  turn) * 10.0f;
    I[1] = fmaxf(0.0f, -turn) * 10.0f;
    I[2] = speed * 5.0f;
    I[3] = fmaxf(0.0f, 0.5f - speed) * 5.0f;
    I[4] = tilt * 8.0f;
    I[5] = tilt * 8.0f;

    float v[T_N], u[T_N], r[T_N];
    #pragma unroll
    for (int j = 0; j < 3; ++j) {
        f32x2 vv = __builtin_nontemporal_load((const f32x2*)(v0_p    + base + 2 * j));
        f32x2 uu = __builtin_nontemporal_load((const f32x2*)(u0_p    + base + 2 * j));
        f32x2 rr = __builtin_nontemporal_load((const f32x2*)(rate0_p + base + 2 * j));
        v[2*j] = vv.x; v[2*j+1] = vv.y;
        u[2*j] = uu.x; u[2*j+1] = uu.y;
        r[2*j] = rr.x; r[2*j+1] = rr.y;
    }

    // ---- Drain the pipeline: step s needs the first 3(s+1) copies done ------
    #pragma unroll
    for (int s = 0; s < T_STEPS; ++s) {
        if      (s == 0) WAIT_ASYNCCNT(27);
        else if (s == 1) WAIT_ASYNCCNT(24);
        else if (s == 2) WAIT_ASYNCCNT(21);
        else if (s == 3) WAIT_ASYNCCNT(18);
        else if (s == 4) WAIT_ASYNCCNT(15);
        else if (s == 5) WAIT_ASYNCCNT(12);
        else if (s == 6) WAIT_ASYNCCNT(9);
        else if (s == 7) WAIT_ASYNCCNT(6);
        else if (s == 8) WAIT_ASYNCCNT(3);
        else             WAIT_ASYNCCNT(0);

        float nz[T_N];
        #pragma unroll
        for (int j = 0; j < 3; ++j) {
            f32x2 nn = *(const f32x2*)&nbuf[s][tid][2 * j];   // ds_load, no bank conflicts
            nz[2*j] = nn.x; nz[2*j+1] = nn.y;
        }

        #pragma unroll
        for (int n = 0; n < T_N; ++n) {
            float i_tot = I[n] + nz[n] * 0.3f + (-1.0f);
            float vv = v[n];
            vv = vv + (0.04f * vv * vv + 5.0f * vv + 140.0f - u[n] + i_tot);
            float uu = u[n] + 0.02f * (0.2f * vv - u[n]);     // uses updated v
            const bool fired = (vv >= 30.0f);
            const float spk = fired ? 1.0f : 0.0f;
            vv = fired ? -65.0f : vv;
            uu = uu + spk * 8.0f;
            r[n] = r[n] * 0.9f + spk * 0.1f;
            v[n] = vv;
            u[n] = uu;
        }
    }

    const float rate_mean =
        (r[0] + r[1] + r[2] + r[3] + r[4] + r[5]) * (1.0f / 6.0f);

    // TwoCompColumn: 8 relaxation substeps, tau = 0.5 (constant-folded)
    float vb0 = 0.0f, vb1 = 0.0f, va0 = 0.0f, va1 = 0.0f;
    #pragma unroll
    for (int s = 0; s < 8; ++s) {
        vb0 += 0.5f * (turn   - vb0);
        vb1 += 0.5f * (speed  - vb1);
        va0 += 0.5f * (pturn  - va0);
        va1 += 0.5f * (pspeed - va1);
    }
    const float pe0 = vb0 - va0;
    const float pe1 = vb1 - va1;
    const float prec0 = 1.0f / (1.0f + pe0 * pe0);
    const float prec1 = 1.0f / (1.0f + pe1 * pe1);
    const float free_energy = 0.5f * (prec0 * pe0 * pe0 + prec1 * pe1 * pe1);
    const float pe_w = 0.7f * pe0 + 0.3f * pe1;
    const float prec_mean = 0.5f * (prec0 + prec1);
    const float postural = -prec_mean * pe_w * 0.3f;

    if (active) {
        // out row: [tilt, rate_mean, postural, pe_w, prec_mean, free_energy]
        f32x2 o0 = {tilt, rate_mean};
        f32x2 o1 = {postural, pe_w};
        f32x2 o2 = {prec_mean, free_energy};
        __builtin_nontemporal_store(o0, (f32x2*)(out_p + base + 0));
        __builtin_nontemporal_store(o1, (f32x2*)(out_p + base + 2));
        __builtin_nontemporal_store(o2, (f32x2*)(out_p + base + 4));
    }
}

extern "C" void kernel_launch(void* const* d_in, const int* in_sizes, int n_in,
                              void* d_out, int out_size, void* d_ws, size_t ws_size,
                              hipStream_t stream) {
    (void)n_in; (void)out_size; (void)d_ws; (void)ws_size;
    // setup_inputs order:
    // 0: heading (unused)  1: speed  2: turn_rate  3: predicted_turn
    // 4: predicted_speed   5: noise [STEPS,B,N]    6: v0  7: u0  8: rate0
    const float* speed_p  = (const float*)d_in[1];
    const float* turn_p   = (const float*)d_in[2];
    const float* pturn_p  = (const float*)d_in[3];
    const float* pspeed_p = (const float*)d_in[4];
    const float* noise_p  = (const float*)d_in[5];
    const float* v0_p     = (const float*)d_in[6];
    const float* u0_p     = (const float*)d_in[7];
    const float* rate0_p  = (const float*)d_in[8];
    float* out_p = (float*)d_out;

    const int B = in_sizes[1];
    const int blocks = (B + THREADS - 1) / THREADS;

    spiking_vestibular_kernel<<<blocks, THREADS, 0, stream>>>(
        speed_p, turn_p, pturn_p, pspeed_p, noise_p, v0_p, u0_p, rate0_p,
        out_p, B);
}